// GCN2_43533788512792
// MI455X (gfx1250) — compile-verified
//
#include <hip/hip_runtime.h>
#include <hip/hip_bf16.h>

#define N_NODES   100000
#define N_EDGES   1600000
#define EMB       16
#define HID       128
#define OUT_DIM   3
#define ALPHA     0.2f
#define NEG_SLOPE 0.01f

typedef float v2f __attribute__((ext_vector_type(2)));
typedef float v8f __attribute__((ext_vector_type(8)));

// ---------------------------------------------------------------------------
// 1. degree accumulation (atomic), norm finalize
// ---------------------------------------------------------------------------
__global__ __launch_bounds__(256) void zero_kernel(float* __restrict__ p, int n) {
    int i = blockIdx.x * 256 + threadIdx.x;
    if (i < n) p[i] = 0.0f;
}

__global__ __launch_bounds__(256) void deg_kernel(const int* __restrict__ ecol,
                                                  float* __restrict__ deg, int nE) {
    int e = blockIdx.x * 256 + threadIdx.x;
    if (e < nE) atomicAdd(&deg[ecol[e]], 1.0f);
}

__global__ __launch_bounds__(256) void dis_kernel(float* __restrict__ deg, int n) {
    int i = blockIdx.x * 256 + threadIdx.x;
    if (i < n) {
        float d = deg[i];
        deg[i] = (d > 0.0f) ? rsqrtf(fmaxf(d, 1.0f)) : 0.0f;  // in place: deg -> dis
    }
}

__global__ __launch_bounds__(256) void norm_kernel(const int* __restrict__ erow,
                                                   const int* __restrict__ ecol,
                                                   const float* __restrict__ dis,
                                                   float* __restrict__ nrm, int nE) {
    int e = blockIdx.x * 256 + threadIdx.x;
    if (e < nE) nrm[e] = dis[erow[e]] * dis[ecol[e]];
}

// ---------------------------------------------------------------------------
// Pre-paired W staging: sWp[(k/2)*HID + col] = { W[k][col], W[k+1][col] }
// so a WMMA B operand is ONE aligned ds_load_b64 into an adjacent VGPR pair.
// ---------------------------------------------------------------------------
__device__ __forceinline__ void stage_w_pairs(v2f* sWp, const float* __restrict__ W,
                                              int K, int tid, int nthreads) {
    const v2f* Wv = (const v2f*)W;            // W row-major [K][HID], rows 512B-aligned
    int total = (K / 2) * (HID / 2);          // each iter handles one (k-pair, col-pair)
    for (int i = tid; i < total; i += nthreads) {
        int p  = i >> 6;                       // k-pair index (HID/2 = 64 col-pairs)
        int c2 = (i & 63) * 2;                 // column (even)
        v2f w0 = Wv[p * HID + (c2 >> 1)];              // W[2p  ][c2], W[2p  ][c2+1]
        v2f w1 = Wv[p * HID + (HID >> 1) + (c2 >> 1)]; // W[2p+1][c2], W[2p+1][c2+1]
        v2f t0; t0.x = w0.x; t0.y = w1.x;
        v2f t1; t1.x = w0.y; t1.y = w1.y;
        sWp[p * HID + c2]     = t0;
        sWp[p * HID + c2 + 1] = t1;
    }
}

// ---------------------------------------------------------------------------
// 2. input projection: x = LeakyReLU(feature @ W_in + b_in)   [N,16]x[16,128]
//    One wave -> 16x128 stripe via V_WMMA_F32_16X16X4_F32, K=16.
// ---------------------------------------------------------------------------
__global__ __launch_bounds__(256)
void input_gemm_kernel(const float* __restrict__ feat,
                       const float* __restrict__ W_in,
                       const float* __restrict__ b_in,
                       float* __restrict__ x, int nRowTiles) {
    __shared__ v2f sWp[(EMB / 2) * HID];  // 8 KB, pre-paired
    stage_w_pairs(sWp, W_in, EMB, threadIdx.x, blockDim.x);
    __syncthreads();

    int wave = threadIdx.x >> 5;
    int lane = threadIdx.x & 31;
    int rowTile = blockIdx.x * 8 + wave;
    if (rowTile >= nRowTiles) return;     // uniform per wave: EXEC all-1 in live waves

    int rowBase = rowTile * 16;
    int m   = lane & 15;                  // M (A) / N (B,D)
    int khp = lane >> 4;                  // k-pair select within k-step of 4
    const v2f* frow = (const v2f*)(feat + (size_t)(rowBase + m) * EMB);

    v8f acc[8];
#pragma unroll
    for (int t = 0; t < 8; ++t) acc[t] = (v8f){0, 0, 0, 0, 0, 0, 0, 0};

#pragma unroll
    for (int kk = 0; kk < EMB; kk += 4) {
        int p0 = (kk >> 1) + khp;         // this lane's k-pair
        v2f a = frow[p0];                 // one global_load_b64
#pragma unroll
        for (int t = 0; t < 8; ++t) {
            v2f b = sWp[p0 * HID + t * 16 + m];   // one ds_load_b64
            acc[t] = __builtin_amdgcn_wmma_f32_16x16x4_f32(
                false, a, false, b, (short)0, acc[t], false, false);
        }
    }

    int halfM = khp * 8;
#pragma unroll
    for (int t = 0; t < 8; ++t) {
        int col = t * 16 + m;
        float bias = b_in[col];
#pragma unroll
        for (int r = 0; r < 8; ++r) {
            float v = acc[t][r] + bias;
            v = (v >= 0.0f) ? v : NEG_SLOPE * v;  // LeakyReLU
            x[(size_t)(rowBase + halfM + r) * HID + col] = v;
        }
    }
}

// ---------------------------------------------------------------------------
// 3. edge aggregation: agg[col] += norm[e] * src[row]   (feature-coalesced)
//    Edge metadata staged into LDS via gfx1250 async copy (ASYNCcnt path).
// ---------------------------------------------------------------------------
#define EPB 64
__global__ __launch_bounds__(128)
void aggregate_kernel(const int* __restrict__ erow, const int* __restrict__ ecol,
                      const float* __restrict__ nrm, const float* __restrict__ src,
                      float* __restrict__ agg, int nE) {
    __shared__ int   sr[EPB];
    __shared__ int   sc[EPB];
    __shared__ float sn[EPB];

    int tid  = threadIdx.x;               // 0..127 = feature lane
    int base = blockIdx.x * EPB;

    if (tid < EPB) {
        // LDS byte offsets (generic LDS-aperture pointer: low 32 bits = LDS addr)
        unsigned ldsR = (unsigned)(size_t)&sr[tid];
        unsigned ldsC = (unsigned)(size_t)&sc[tid];
        unsigned ldsN = (unsigned)(size_t)&sn[tid];
        unsigned vofs = (unsigned)(base + tid) * 4u;   // byte offset in stream
        asm volatile("global_load_async_to_lds_b32 %0, %1, %2"
                     :: "v"(ldsR), "v"(vofs), "s"(erow) : "memory");
        asm volatile("global_load_async_to_lds_b32 %0, %1, %2"
                     :: "v"(ldsC), "v"(vofs), "s"(ecol) : "memory");
        asm volatile("global_load_async_to_lds_b32 %0, %1, %2"
                     :: "v"(ldsN), "v"(vofs), "s"(nrm) : "memory");
        // prefetch next block's edge metadata while we chew on this one
        __builtin_prefetch(&erow[base + EPB + tid], 0, 1);
    }
    asm volatile("s_wait_asynccnt 0" ::: "memory");
    __syncthreads();

    int cnt = nE - base;
    if (cnt > EPB) cnt = EPB;
    for (int i = 0; i < cnt; ++i) {
        int   r  = sr[i];
        int   c  = sc[i];
        float nv = sn[i];
        atomicAdd(&agg[(size_t)c * HID + tid], nv * src[(size_t)r * HID + tid]);
    }
}

// ---------------------------------------------------------------------------
// 4. GCN2 layer GEMM: out = ((1-a)*agg + a*x0) @ W    [N,128]x[128,128]
//    W pre-paired in LDS (64 KB of 320 KB/WGP), one wave -> 16x128 stripe.
// ---------------------------------------------------------------------------
__global__ __launch_bounds__(256)
void gcn_gemm_kernel(const float* __restrict__ agg, const float* __restrict__ x0,
                     const float* __restrict__ W, float* __restrict__ out,
                     int nRowTiles) {
    __shared__ v2f sWp[(HID / 2) * HID];  // 64 KB, pre-paired
    stage_w_pairs(sWp, W, HID, threadIdx.x, blockDim.x);
    __syncthreads();

    int wave = threadIdx.x >> 5;
    int lane = threadIdx.x & 31;
    int rowTile = blockIdx.x * 8 + wave;
    if (rowTile >= nRowTiles) return;     // uniform per wave

    int rowBase = rowTile * 16;
    int m   = lane & 15;
    int khp = lane >> 4;
    const v2f* arow = (const v2f*)(agg + (size_t)(rowBase + m) * HID);
    const v2f* xrow = (const v2f*)(x0  + (size_t)(rowBase + m) * HID);

    v8f acc[8];
#pragma unroll
    for (int t = 0; t < 8; ++t) acc[t] = (v8f){0, 0, 0, 0, 0, 0, 0, 0};

    for (int kk = 0; kk < HID; kk += 4) {
        int p0 = (kk >> 1) + khp;
        // residual mix fused into the vector A load (2x global_load_b64 + v_fma)
        v2f a = arow[p0] * (1.0f - ALPHA) + xrow[p0] * ALPHA;
#pragma unroll
        for (int t = 0; t < 8; ++t) {
            v2f b = sWp[p0 * HID + t * 16 + m];   // one ds_load_b64, no re-marshal
            acc[t] = __builtin_amdgcn_wmma_f32_16x16x4_f32(
                false, a, false, b, (short)0, acc[t], false, false);
        }
    }

    int halfM = khp * 8;
#pragma unroll
    for (int t = 0; t < 8; ++t) {
        int col = t * 16 + m;
#pragma unroll
        for (int r = 0; r < 8; ++r)
            out[(size_t)(rowBase + halfM + r) * HID + col] = acc[t][r];
    }
}

// ---------------------------------------------------------------------------
// 5. output projection: out = h @ W_out + b_out   [N,128]x[128,3]
//    One wave per node, lane-parallel dot + wave32 shuffle reduction.
// ---------------------------------------------------------------------------
__global__ __launch_bounds__(256)
void out_gemm_kernel(const float* __restrict__ h, const float* __restrict__ W_out,
                     const float* __restrict__ b_out, float* __restrict__ out,
                     int nNodes) {
    int wave = threadIdx.x >> 5;
    int lane = threadIdx.x & 31;
    int node = blockIdx.x * 8 + wave;
    if (node >= nNodes) return;

    float v0 = h[(size_t)node * HID + lane];
    float v1 = h[(size_t)node * HID + lane + 32];
    float v2 = h[(size_t)node * HID + lane + 64];
    float v3 = h[(size_t)node * HID + lane + 96];

#pragma unroll
    for (int o = 0; o < OUT_DIM; ++o) {
        float p = v0 * W_out[lane * OUT_DIM + o]
                + v1 * W_out[(lane + 32) * OUT_DIM + o]
                + v2 * W_out[(lane + 64) * OUT_DIM + o]
                + v3 * W_out[(lane + 96) * OUT_DIM + o];
#pragma unroll
        for (int off = 16; off >= 1; off >>= 1)
            p += __shfl_down(p, off, 32);
        if (lane == 0) out[(size_t)node * OUT_DIM + o] = p + b_out[o];
    }
}

// ---------------------------------------------------------------------------
// launch
// ---------------------------------------------------------------------------
extern "C" void kernel_launch(void* const* d_in, const int* in_sizes, int n_in,
                              void* d_out, int out_size, void* d_ws, size_t ws_size,
                              hipStream_t stream) {
    const float* feature = (const float*)d_in[0];
    const int*   eidx    = (const int*)d_in[1];     // [2, E]
    // d_in[2] edge_type: unused by the reference computation
    const float* W_in    = (const float*)d_in[3];
    const float* b_in    = (const float*)d_in[4];
    const float* W_g1    = (const float*)d_in[5];
    const float* W_g2    = (const float*)d_in[6];
    const float* W_out   = (const float*)d_in[7];
    const float* b_out   = (const float*)d_in[8];
    float* out = (float*)d_out;

    const int* erow = eidx;
    const int* ecol = eidx + N_EDGES;

    // workspace layout (floats): dis[N] | norm[E] | x[N*H] | agg[N*H] | h[N*H]
    float* dis = (float*)d_ws;
    float* nrm = dis + N_NODES;
    float* x   = nrm + N_EDGES;
    float* agg = x + (size_t)N_NODES * HID;
    float* h   = agg + (size_t)N_NODES * HID;

    const int NH = N_NODES * HID;
    const int nRowTiles = N_NODES / 16;                        // 6250 exact
    const dim3 gemmGrid((nRowTiles + 7) / 8);
    const dim3 aggGrid((N_EDGES + EPB - 1) / EPB);

    // --- normalization ---
    zero_kernel<<<(N_NODES + 255) / 256, 256, 0, stream>>>(dis, N_NODES);
    deg_kernel<<<(N_EDGES + 255) / 256, 256, 0, stream>>>(ecol, dis, N_EDGES);
    dis_kernel<<<(N_NODES + 255) / 256, 256, 0, stream>>>(dis, N_NODES);
    norm_kernel<<<(N_EDGES + 255) / 256, 256, 0, stream>>>(erow, ecol, dis, nrm, N_EDGES);

    // --- input projection + LeakyReLU ---
    input_gemm_kernel<<<gemmGrid, 256, 0, stream>>>(feature, W_in, b_in, x, nRowTiles);

    // --- layer 1: aggregate(x) -> residual -> @ W_g1 ---
    zero_kernel<<<(NH + 255) / 256, 256, 0, stream>>>(agg, NH);
    aggregate_kernel<<<aggGrid, 128, 0, stream>>>(erow, ecol, nrm, x, agg, N_EDGES);
    gcn_gemm_kernel<<<gemmGrid, 256, 0, stream>>>(agg, x, W_g1, h, nRowTiles);

    // --- layer 2: aggregate(h) -> residual -> @ W_g2 (h overwritten) ---
    zero_kernel<<<(NH + 255) / 256, 256, 0, stream>>>(agg, NH);
    aggregate_kernel<<<aggGrid, 128, 0, stream>>>(erow, ecol, nrm, h, agg, N_EDGES);
    gcn_gemm_kernel<<<gemmGrid, 256, 0, stream>>>(agg, x, W_g2, h, nRowTiles);

    // --- output projection ---
    out_gemm_kernel<<<(N_NODES + 7) / 8, 256, 0, stream>>>(h, W_out, b_out, out, N_NODES);
}